// STA_LSTM_67697274519659
// MI455X (gfx1250) — compile-verified
//
#include <hip/hip_runtime.h>

// ============================================================================
// STA-LSTM for MI455X (gfx1250), fp32 via V_WMMA_F32_16X16X4_F32.
//
// Input index assumption (setup_inputs() insertion order, leaves in dict order):
//  0:X 1:Y
//  tcn0: 2:v1 3:g1 4:b1 5:v2 6:g2 7:b2
//  tcn1: 8:v1 9:g1 10:b1 11:v2 12:g2 13:b2 14:dw 15:db
//  tcn2: 16:v1 17:g1 18:b1 19:v2 20:g2 21:b2 22:dw 23:db
//  tcn3: 24:v1 25:g1 26:b1 27:v2 28:g2 29:b2
//  lstm: 30:Wih 31:Whh 32:bih 33:bhh   fc: 34:w 35:b
//  ta:   36:Wa 37:Ua 38:ba 39:Va 40:W 41:U 42:bias 43:Wy 44:fcw 45:fcb
//  46: epoch (unused)
// ============================================================================

typedef float v2f __attribute__((ext_vector_type(2)));
typedef float v8f __attribute__((ext_vector_type(8)));

__device__ __forceinline__ v8f wmma4(v2f a, v2f b, v8f c) {
  // D(16x16,f32) = A(16x4,f32) * B(4x16,f32) + C
  return __builtin_amdgcn_wmma_f32_16x16x4_f32(false, a, false, b, (short)0, c,
                                               false, false);
}
__device__ __forceinline__ float sigm(float x) { return 1.0f / (1.0f + __expf(-x)); }

// Model constants
#define NB   512      // batch
#define NL   128      // seq len
#define NBL  65536    // NB*NL
#define XC   128      // channels of X (F_APP + F)
#define FAPP 32
#define FCH  96
#define SAH  256
#define G4   1024     // 4*SA_H
#define TOUT 24
#define NTN  64
#define TAH  256

// ---------------------------------------------------------------------------
// Generic GEMM:  Cout[n, m] = act( sum_k A[m,k]*B[n,k] + bias[m] + Cin[n,m] )
// A element address: A + m*am + k*ak (supports strided "transposed" weights).
// B is row-major (n rows, k contiguous, leading dim ldb).
// Wave tile: M=16, N=64 (4 WMMA accumulators). Block = 4 waves = 64 M rows.
// ---------------------------------------------------------------------------
__global__ __launch_bounds__(128) void gemm_wmma(
    const float* __restrict__ A, long am, long ak,
    const float* __restrict__ B, long ldb,
    const float* __restrict__ bias,
    const float* __restrict__ Cin, long cld,
    float* __restrict__ Cout, long ldc,
    int M, int N, int K, int act) {
  int wave = threadIdx.x >> 5;
  int lane = threadIdx.x & 31;
  int lrow = lane & 15, lhi = lane >> 4;
  int m0 = blockIdx.y * 64 + wave * 16;
  if (m0 >= M) return;
  int n0 = blockIdx.x * 64;

  const float* arow = A + (long)(m0 + lrow) * am;
  const float* brow[4];
#pragma unroll
  for (int s = 0; s < 4; s++) brow[s] = B + (long)(n0 + 16 * s + lrow) * ldb;

  v8f acc[4] = {};
  for (int k0 = 0; k0 < K; k0 += 4) {
    int kk = k0 + 2 * lhi;
    v2f a;
    a.x = arow[(long)kk * ak];
    a.y = arow[(long)(kk + 1) * ak];
#pragma unroll
    for (int s = 0; s < 4; s++) {
      float2 bv = *(const float2*)(brow[s] + kk);
      v2f b; b.x = bv.x; b.y = bv.y;
      acc[s] = wmma4(a, b, acc[s]);
    }
  }

  int mb = m0 + lhi * 8;
  float bvv[8];
#pragma unroll
  for (int r = 0; r < 8; r++) bvv[r] = bias ? bias[mb + r] : 0.0f;
#pragma unroll
  for (int s = 0; s < 4; s++) {
    int n = n0 + 16 * s + lrow;
    float* op = Cout + (long)n * ldc + mb;
    const float* ci = Cin ? (Cin + (long)n * cld + mb) : nullptr;
#pragma unroll
    for (int r = 0; r < 8; r++) {
      float v = acc[s][r] + bvv[r];
      if (ci) v += ci[r];
      if (act) v = fmaxf(v, 0.0f);
      op[r] = v;
    }
  }
}

// ---------------------------------------------------------------------------
// Causal dilated conv, K=2, as GEMM over rows n = b*128 + t:
//   out[n,o] = relu( W0[o,:]*x[n-d,:] (if t>=d) + W1[o,:]*x[n,:] + bias[o] )
//   if Res: out = relu(out + Res[n,:])           (matches relu(relu(conv)+res))
// ---------------------------------------------------------------------------
__global__ __launch_bounds__(128) void conv_k2_wmma(
    const float* __restrict__ W0, const float* __restrict__ W1, int I,
    const float* __restrict__ Xin, int ldx, int d,
    const float* __restrict__ bias,
    const float* __restrict__ Res, int rld,
    float* __restrict__ Out, int M, int N) {
  int wave = threadIdx.x >> 5;
  int lane = threadIdx.x & 31;
  int lrow = lane & 15, lhi = lane >> 4;
  int m0 = blockIdx.y * 64 + wave * 16;
  if (m0 >= M) return;
  int n0 = blockIdx.x * 64;

  const float* a0row = W0 + (long)(m0 + lrow) * I;
  const float* a1row = W1 + (long)(m0 + lrow) * I;
  const float* bx1[4];
  const float* bx0[4];
  bool ok0[4];
#pragma unroll
  for (int s = 0; s < 4; s++) {
    int n = n0 + 16 * s + lrow;
    bx1[s] = Xin + (long)n * ldx;
    ok0[s] = (n & 127) >= d;            // causal zero-pad guard (L=128)
    bx0[s] = Xin + (long)(n - d) * ldx;
  }

  v8f acc[4] = {};
  for (int k0 = 0; k0 < I; k0 += 4) {
    int kk = k0 + 2 * lhi;
    float2 av0 = *(const float2*)(a0row + kk);
    float2 av1 = *(const float2*)(a1row + kk);
    v2f a0; a0.x = av0.x; a0.y = av0.y;
    v2f a1; a1.x = av1.x; a1.y = av1.y;
#pragma unroll
    for (int s = 0; s < 4; s++) {
      float2 b1v = *(const float2*)(bx1[s] + kk);
      v2f b1; b1.x = b1v.x; b1.y = b1v.y;
      acc[s] = wmma4(a1, b1, acc[s]);
      v2f b0; b0.x = 0.0f; b0.y = 0.0f;
      if (ok0[s]) {
        float2 b0v = *(const float2*)(bx0[s] + kk);
        b0.x = b0v.x; b0.y = b0v.y;
      }
      acc[s] = wmma4(a0, b0, acc[s]);
    }
  }

  int mb = m0 + lhi * 8;
  float bvv[8];
#pragma unroll
  for (int r = 0; r < 8; r++) bvv[r] = bias[mb + r];
#pragma unroll
  for (int s = 0; s < 4; s++) {
    int n = n0 + 16 * s + lrow;
    float* op = Out + (long)n * M + mb;
    const float* rp = Res ? (Res + (long)n * rld + mb) : nullptr;
#pragma unroll
    for (int r = 0; r < 8; r++) {
      float v = fmaxf(acc[s][r] + bvv[r], 0.0f);
      if (rp) v = fmaxf(v + rp[r], 0.0f);
      op[r] = v;
    }
  }
}

// ---------------------------------------------------------------------------
// Fused LSTM gate GEMM: Z[b,m] = x_t[b,:]@Wih[m,:] + h[b,:]@Whh[m,:] + bias2[m]
// x_t[b,k] = X[b*16384 + t*128 + k] (first 32 channels).  M=1024, N=512.
// ---------------------------------------------------------------------------
__global__ __launch_bounds__(128) void lstm_gates_wmma(
    const float* __restrict__ Wih, const float* __restrict__ Whh,
    const float* __restrict__ X, int t,
    const float* __restrict__ h, const float* __restrict__ bias2,
    float* __restrict__ Z) {
  int wave = threadIdx.x >> 5;
  int lane = threadIdx.x & 31;
  int lrow = lane & 15, lhi = lane >> 4;
  int m0 = blockIdx.y * 64 + wave * 16;
  int n0 = blockIdx.x * 64;

  const float* axr = Wih + (long)(m0 + lrow) * FAPP;
  const float* ahr = Whh + (long)(m0 + lrow) * SAH;
  const float* bx[4];
  const float* bh[4];
#pragma unroll
  for (int s = 0; s < 4; s++) {
    int n = n0 + 16 * s + lrow;
    bx[s] = X + (long)n * (NL * XC) + (long)t * XC;
    bh[s] = h + (long)n * SAH;
  }

  v8f acc[4] = {};
  for (int k0 = 0; k0 < FAPP; k0 += 4) {   // x segment (K=32)
    int kk = k0 + 2 * lhi;
    float2 av = *(const float2*)(axr + kk);
    v2f a; a.x = av.x; a.y = av.y;
#pragma unroll
    for (int s = 0; s < 4; s++) {
      float2 bv = *(const float2*)(bx[s] + kk);
      v2f b; b.x = bv.x; b.y = bv.y;
      acc[s] = wmma4(a, b, acc[s]);
    }
  }
  for (int k0 = 0; k0 < SAH; k0 += 4) {    // h segment (K=256)
    int kk = k0 + 2 * lhi;
    float2 av = *(const float2*)(ahr + kk);
    v2f a; a.x = av.x; a.y = av.y;
#pragma unroll
    for (int s = 0; s < 4; s++) {
      float2 bv = *(const float2*)(bh[s] + kk);
      v2f b; b.x = bv.x; b.y = bv.y;
      acc[s] = wmma4(a, b, acc[s]);
    }
  }

  int mb = m0 + lhi * 8;
#pragma unroll
  for (int s = 0; s < 4; s++) {
    int n = n0 + 16 * s + lrow;
    float* op = Z + (long)n * G4 + mb;
#pragma unroll
    for (int r = 0; r < 8; r++) op[r] = acc[s][r] + bias2[mb + r];
  }
}

// ---------------------------------------------------------------------------
// Temporal gate GEMM: Z[b,m] = hctx[b,:96]@W[:,t,m] + h[b,:256]@U[:,t,m] + bias[t,m]
// W: (96,24,1024)  U: (256,24,1024) -> A[m,k] at base + m + k*24576
// ---------------------------------------------------------------------------
__global__ __launch_bounds__(128) void ta_gates_wmma(
    const float* __restrict__ Wfull, const float* __restrict__ Ufull, int t,
    const float* __restrict__ hctx, const float* __restrict__ h,
    const float* __restrict__ biasfull, float* __restrict__ Z) {
  int wave = threadIdx.x >> 5;
  int lane = threadIdx.x & 31;
  int lrow = lane & 15, lhi = lane >> 4;
  int m0 = blockIdx.y * 64 + wave * 16;
  int n0 = blockIdx.x * 64;

  const long KS = (long)TOUT * G4;  // 24576
  const float* Wt = Wfull + (long)t * G4 + (m0 + lrow);
  const float* Ut = Ufull + (long)t * G4 + (m0 + lrow);
  const float* bc[4];
  const float* bh[4];
#pragma unroll
  for (int s = 0; s < 4; s++) {
    int n = n0 + 16 * s + lrow;
    bc[s] = hctx + (long)n * FCH;
    bh[s] = h + (long)n * SAH;
  }

  v8f acc[4] = {};
  for (int k0 = 0; k0 < FCH; k0 += 4) {    // hctx segment (K=96)
    int kk = k0 + 2 * lhi;
    v2f a; a.x = Wt[(long)kk * KS]; a.y = Wt[(long)(kk + 1) * KS];
#pragma unroll
    for (int s = 0; s < 4; s++) {
      float2 bv = *(const float2*)(bc[s] + kk);
      v2f b; b.x = bv.x; b.y = bv.y;
      acc[s] = wmma4(a, b, acc[s]);
    }
  }
  for (int k0 = 0; k0 < SAH; k0 += 4) {    // h segment (K=256)
    int kk = k0 + 2 * lhi;
    v2f a; a.x = Ut[(long)kk * KS]; a.y = Ut[(long)(kk + 1) * KS];
#pragma unroll
    for (int s = 0; s < 4; s++) {
      float2 bv = *(const float2*)(bh[s] + kk);
      v2f b; b.x = bv.x; b.y = bv.y;
      acc[s] = wmma4(a, b, acc[s]);
    }
  }

  const float* bias = biasfull + (long)t * G4;
  int mb = m0 + lhi * 8;
#pragma unroll
  for (int s = 0; s < 4; s++) {
    int n = n0 + 16 * s + lrow;
    float* op = Z + (long)n * G4 + mb;
#pragma unroll
    for (int r = 0; r < 8; r++) op[r] = acc[s][r] + bias[mb + r];
  }
}

// ---------------------------------------------------------------------------
// Elementwise / reduction kernels
// ---------------------------------------------------------------------------
__global__ void wn_prep(const float* __restrict__ v, const float* __restrict__ g,
                        float* __restrict__ W0, float* __restrict__ W1, int I) {
  int o = blockIdx.x, tid = threadIdx.x;
  const float* vr = v + (long)o * I * 2;
  float ss = 0.0f;
  for (int i = tid; i < I * 2; i += 128) { float x = vr[i]; ss += x * x; }
  __shared__ float red[128];
  red[tid] = ss; __syncthreads();
  for (int s = 64; s > 0; s >>= 1) {
    if (tid < s) red[tid] += red[tid + s];
    __syncthreads();
  }
  float scale = g[o] * rsqrtf(red[0]);
  for (int i = tid; i < I; i += 128) {
    W0[(long)o * I + i] = vr[2 * i] * scale;      // tap kk=0 -> x[t-d]
    W1[(long)o * I + i] = vr[2 * i + 1] * scale;  // tap kk=1 -> x[t]
  }
}

__global__ void misc_init(const float* bih, const float* bhh, float* bias2,
                          float* h, float* c) {
  int i = blockIdx.x * 256 + threadIdx.x;
  if (i < G4) bias2[i] = bih[i] + bhh[i];
  if (i < NB * SAH) { h[i] = 0.0f; c[i] = 0.0f; }
}

__global__ void lstm_cell(const float* __restrict__ Z, float* __restrict__ h,
                          float* __restrict__ c) {
  int b = blockIdx.x, u = threadIdx.x;
  const float* z = Z + (long)b * G4;
  float cc = c[b * SAH + u];
  cc = sigm(z[SAH + u]) * cc + sigm(z[u]) * tanhf(z[2 * SAH + u]);
  float hh = sigm(z[3 * SAH + u]) * tanhf(cc);
  c[b * SAH + u] = cc;
  h[b * SAH + u] = hh;
}

__global__ void ta_init(const float* emb, const float* X, float* h, float* c,
                        float* y) {
  int b = blockIdx.x, u = threadIdx.x;
  h[b * SAH + u] = emb[b * 512 + u];
  c[b * SAH + u] = emb[b * 512 + SAH + u];
  if (u == 0) y[b] = X[(long)b * (NL * XC) + (NL * XC - 1)];  // X[b,-1,-1]
}

// attention: e=tanh(Pt + cUa(+ba)) ; s=e.Va/16 ; softmax over L ; hctx=beta@Hmap
__global__ void attn_kernel(const float* __restrict__ Pt,
                            const float* __restrict__ cua,
                            const float* __restrict__ Va, int t,
                            const float* __restrict__ Hmap,
                            float* __restrict__ hctx) {
  int b = blockIdx.x, l = threadIdx.x;
  const float* p = Pt + (long)(b * NL + l) * NTN;
  const float* cu = cua + b * NTN;
  float s = 0.0f;
  for (int n = 0; n < NTN; n++) {
    float e = tanhf(p[n] + cu[n]);
    s += e * Va[n * TOUT + t];
  }
  s *= 0.0625f;  // 1/sqrt(256)
  __shared__ float sb[NL], eb[NL];
  sb[l] = s; __syncthreads();
  float m = -1e30f;
  for (int j = 0; j < NL; j++) m = fmaxf(m, sb[j]);
  float e = __expf(s - m);
  eb[l] = e; __syncthreads();
  float sum = 0.0f;
  for (int j = 0; j < NL; j++) sum += eb[j];
  float beta = e / sum;
  sb[l] = beta; __syncthreads();
  if (l < FCH) {
    const float* hm = Hmap + (long)b * NL * FCH + l;
    float acc = 0.0f;
    for (int j = 0; j < NL; j++) acc += sb[j] * hm[(long)j * FCH];
    hctx[b * FCH + l] = acc;
  }
}

// temporal cell: gates += y*Wy_t ; LSTM update ; y_new = h.fcw_t + fcb_t
__global__ void ta_cell(const float* __restrict__ Z, const float* __restrict__ Wy,
                        int t, float* __restrict__ y, float* __restrict__ h,
                        float* __restrict__ c, const float* __restrict__ fcw,
                        const float* __restrict__ fcb, float* __restrict__ out) {
  int b = blockIdx.x, u = threadIdx.x;
  float yb = y[b];
  const float* z = Z + (long)b * G4;
  const float* wy = Wy + (long)t * G4;
  float zi = z[u] + yb * wy[u];
  float zf = z[SAH + u] + yb * wy[SAH + u];
  float zg = z[2 * SAH + u] + yb * wy[2 * SAH + u];
  float zo = z[3 * SAH + u] + yb * wy[3 * SAH + u];
  float cc = c[b * SAH + u];
  cc = sigm(zf) * cc + sigm(zi) * tanhf(zg);
  float hh = sigm(zo) * tanhf(cc);
  c[b * SAH + u] = cc;
  h[b * SAH + u] = hh;
  __shared__ float red[SAH];
  red[u] = hh * fcw[t * SAH + u];
  __syncthreads();
  for (int s = 128; s > 0; s >>= 1) {
    if (u < s) red[u] += red[u + s];
    __syncthreads();
  }
  if (u == 0) {
    float yn = red[0] + fcb[t];
    y[b] = yn;
    out[(long)b * TOUT + t] = yn;
  }
}

// ---------------------------------------------------------------------------
extern "C" void kernel_launch(void* const* d_in, const int* in_sizes, int n_in,
                              void* d_out, int out_size, void* d_ws,
                              size_t ws_size, hipStream_t stream) {
  (void)in_sizes; (void)n_in; (void)out_size; (void)ws_size;
  auto in = [&](int i) { return (const float*)d_in[i]; };

  const float* X = in(0);
  // TCN conv params (order: b0c1,b0c2,b1c1,b1c2,b2c1,b2c2,b3c1,b3c2)
  const float* cv[8] = {in(2), in(5), in(8), in(11), in(16), in(19), in(24), in(27)};
  const float* cg[8] = {in(3), in(6), in(9), in(12), in(17), in(20), in(25), in(28)};
  const float* cb[8] = {in(4), in(7), in(10), in(13), in(18), in(21), in(26), in(29)};
  const int cO[8] = {96, 96, 192, 192, 96, 96, 96, 96};
  const int cI[8] = {96, 96, 96, 192, 192, 96, 96, 96};
  const float* dw1 = in(14); const float* db1 = in(15);
  const float* dw2 = in(22); const float* db2 = in(23);
  const float* Wih = in(30); const float* Whh = in(31);
  const float* bih = in(32); const float* bhh = in(33);
  const float* fcw2 = in(34); const float* fcb2 = in(35);
  const float* Wa = in(36); const float* Ua = in(37); const float* ba = in(38);
  const float* Va = in(39); const float* Wt = in(40); const float* Ut = in(41);
  const float* tab = in(42); const float* Wy = in(43);
  const float* tfcw = in(44); const float* tfcb = in(45);
  float* out = (float*)d_out;

  // ---- workspace layout (~198 MB of f32) ----
  float* ws = (float*)d_ws;
  size_t off = 0;
  auto alloc = [&](size_t n) { float* p = ws + off; off += n; return p; };
  float* W0[8]; float* W1[8];
  for (int j = 0; j < 8; j++) {
    W0[j] = alloc((size_t)cO[j] * cI[j]);
    W1[j] = alloc((size_t)cO[j] * cI[j]);
  }
  float* bias2 = alloc(G4);
  float* buf1 = alloc((size_t)NBL * 192);
  float* buf2 = alloc((size_t)NBL * 192);
  float* buf3 = alloc((size_t)NBL * 192);
  float* Hmap = alloc((size_t)NBL * FCH);
  float* hbuf = alloc((size_t)NB * SAH);
  float* cbuf = alloc((size_t)NB * SAH);
  float* Z = alloc((size_t)NB * G4);
  float* emb = alloc((size_t)NB * 512);
  float* Pt = alloc((size_t)NBL * NTN);
  float* cua = alloc((size_t)NB * NTN);
  float* hctx = alloc((size_t)NB * FCH);
  float* ybuf = alloc(NB);

  auto gemm = [&](const float* A, long am, long ak, const float* B, long ldb,
                  const float* bias, const float* Cin, long cld, float* C,
                  long ldc, int M, int N, int K, int act) {
    gemm_wmma<<<dim3(N / 64, (M + 63) / 64), 128, 0, stream>>>(
        A, am, ak, B, ldb, bias, Cin, cld, C, ldc, M, N, K, act);
  };
  auto conv = [&](int j, const float* Xin, int ldx, int d, const float* Res,
                  int rld, float* Out) {
    conv_k2_wmma<<<dim3(NBL / 64, (cO[j] + 63) / 64), 128, 0, stream>>>(
        W0[j], W1[j], cI[j], Xin, ldx, d, cb[j], Res, rld, Out, cO[j], NBL);
  };

  // ---- prep: weight-norm taps, fused LSTM bias, zero h/c ----
  for (int j = 0; j < 8; j++)
    wn_prep<<<cO[j], 128, 0, stream>>>(cv[j], cg[j], W0[j], W1[j], cI[j]);
  misc_init<<<(NB * SAH) / 256, 256, 0, stream>>>(bih, bhh, bias2, hbuf, cbuf);

  // ---- TCN (input view: X + 32, ld 128) ----
  const float* Xin0 = X + FAPP;
  conv(0, Xin0, XC, 1, nullptr, 0, buf3);          // b0 conv1 -> buf3
  conv(1, buf3, 96, 1, Xin0, XC, buf1);            // b0 conv2 + id res -> buf1
  gemm(dw1, 96, 1, buf1, 96, db1, nullptr, 0, buf2, 192, 192, NBL, 96, 0);
  conv(2, buf1, 96, 2, nullptr, 0, buf3);          // b1 conv1 -> buf3 (192ch)
  conv(3, buf3, 192, 2, buf2, 192, buf2);          // b1 conv2 + dw res -> buf2
  gemm(dw2, 192, 1, buf2, 192, db2, nullptr, 0, buf1, 96, 96, NBL, 192, 0);
  conv(4, buf2, 192, 4, nullptr, 0, buf3);         // b2 conv1 -> buf3 (96ch)
  conv(5, buf3, 96, 4, buf1, 96, buf1);            // b2 conv2 + dw res -> buf1
  conv(6, buf1, 96, 8, nullptr, 0, buf3);          // b3 conv1 -> buf3
  conv(7, buf3, 96, 8, buf1, 96, Hmap);            // b3 conv2 + id res -> Hmap

  // ---- spatial LSTM (128 sequential steps) ----
  for (int t = 0; t < NL; t++) {
    lstm_gates_wmma<<<dim3(NB / 64, G4 / 64), 128, 0, stream>>>(
        Wih, Whh, X, t, hbuf, bias2, Z);
    lstm_cell<<<NB, SAH, 0, stream>>>(Z, hbuf, cbuf);
  }

  // ---- emb = relu(h @ fc.w^T + fc.b); split into (h0, c0); y0 ----
  gemm(fcw2, SAH, 1, hbuf, SAH, fcb2, nullptr, 0, emb, 512, 512, NB, SAH, 1);
  ta_init<<<NB, SAH, 0, stream>>>(emb, X, hbuf, cbuf, ybuf);

  // ---- temporal attention LSTM (24 sequential steps) ----
  for (int t = 0; t < TOUT; t++) {
    // Pt[(b,l),n] = Hmap[(b,l),:] @ Wa[:,t,n]
    gemm(Wa + (long)t * NTN, 1, (long)TOUT * NTN, Hmap, FCH, nullptr, nullptr, 0,
         Pt, NTN, NTN, NBL, FCH, 0);
    // cua[b,n] = c @ Ua[:,t,n] + ba[t,n]
    gemm(Ua + (long)t * NTN, 1, (long)TOUT * NTN, cbuf, SAH, ba + (long)t * NTN,
         nullptr, 0, cua, NTN, NTN, NB, SAH, 0);
    attn_kernel<<<NB, NL, 0, stream>>>(Pt, cua, Va, t, Hmap, hctx);
    ta_gates_wmma<<<dim3(NB / 64, G4 / 64), 128, 0, stream>>>(
        Wt, Ut, t, hctx, hbuf, tab, Z);
    ta_cell<<<NB, SAH, 0, stream>>>(Z, Wy, t, ybuf, hbuf, cbuf, tfcw, tfcb, out);
  }
}